// AdvancedQuantumCircuitLayer_38620345926119
// MI455X (gfx1250) — compile-verified
//
#include <hip/hip_runtime.h>

typedef __attribute__((ext_vector_type(2))) float v2f;
typedef __attribute__((ext_vector_type(8))) float v8f;

#define NQ      12
#define DIM     4096
#define NLAYERS 6
#define NMAT    21          // 7 stages (6 layers + readout) x 3 blocks

// D(16x16 f32) = A(16x4 f32) * B(4x16 f32) + C   -- chained 4x for K=16
__device__ __forceinline__ v8f wmma_f32(v2f a, v2f b, v8f c) {
  return __builtin_amdgcn_wmma_f32_16x16x4_f32(
      /*neg_a=*/false, a, /*neg_b=*/false, b,
      /*c_mod=*/(short)0, c, /*reuse_a=*/false, /*reuse_b=*/false);
}

// Flat LDS index of state-matrix element (block-row j in 0..15, column r in 0..255)
// for 4-qubit block b. Qubit q occupies bit (11-q) of the flat index.
__device__ __forceinline__ int saddr(int b, int j, int r) {
  if (b == 0)       return (j << 8) | r;                              // bits 11..8
  else if (b == 1)  return ((r >> 4) << 8) | (j << 4) | (r & 15);     // bits 7..4
  else              return (r << 4) | j;                              // bits 3..0
}

// ---------------------------------------------------------------------------
// Prep: build the 21 fused 16x16 complex block unitaries.
// Stage l<6, block b: Kron(RZ*RY*RX for qubits 4b..4b+3), angles = w*scale[l].
// Stage l==6: readout RY layer (real).
// One 256-thread block per matrix; thread t computes entry (t>>4, t&15).
// ---------------------------------------------------------------------------
__global__ void qc_prep(const float* __restrict__ w,
                        const float* __restrict__ scal,
                        float* __restrict__ Mre, float* __restrict__ Mim) {
  const int mat = blockIdx.x;          // l*3 + b
  const int l = mat / 3;
  const int b = mat - 3 * l;
  const int t = threadIdx.x;
  const int r = t >> 4;
  const int c = t & 15;
  float pr = 1.0f, pi = 0.0f;
  #pragma unroll
  for (int k = 0; k < 4; ++k) {        // k=0 is block MSB -> qubit 4b+k
    const int q  = 4 * b + k;
    const int rb = (r >> (3 - k)) & 1;
    const int cb = (c >> (3 - k)) & 1;
    float fre, fim;
    if (l < NLAYERS) {
      const float sc = scal[l];
      const float tx = w[l * 36 + q] * sc;
      const float ty = w[l * 36 + 12 + q] * sc;
      const float tz = w[l * 36 + 24 + q] * sc;
      const float cx = cosf(0.5f * tx), sx = sinf(0.5f * tx);
      const float cy = cosf(0.5f * ty), sy = sinf(0.5f * ty);
      const float cz = cosf(0.5f * tz), sz = sinf(0.5f * tz);
      // M1 = RY*RX
      float m1re, m1im;
      if (rb == 0) {
        if (cb == 0) { m1re =  cy * cx; m1im =  sy * sx; }
        else         { m1re = -sy * cx; m1im = -cy * sx; }
      } else {
        if (cb == 0) { m1re =  sy * cx; m1im = -cy * sx; }
        else         { m1re =  cy * cx; m1im = -sy * sx; }
      }
      // RZ phase on row rb: e^{-i tz/2} (rb=0) or e^{+i tz/2} (rb=1)
      const float pzre = cz;
      const float pzim = (rb == 0) ? -sz : sz;
      fre = pzre * m1re - pzim * m1im;
      fim = pzre * m1im + pzim * m1re;
    } else {
      const float ty = w[216 + q];     // readout RY
      const float cy = cosf(0.5f * ty), sy = sinf(0.5f * ty);
      fre = (rb == 0) ? ((cb == 0) ? cy : -sy)
                      : ((cb == 0) ? sy :  cy);
      fim = 0.0f;
    }
    const float nr = pr * fre - pi * fim;
    const float ni = pr * fim + pi * fre;
    pr = nr; pi = ni;
  }
  Mre[mat * 256 + r * 16 + c] = pr;
  Mim[mat * 256 + r * 16 + c] = pi;
}

// ---------------------------------------------------------------------------
// Main: one 256-thread workgroup (8 wave32) per batch sample.
// State (re/im f32, 4096 each) lives in 32KB LDS for the whole circuit.
// Per stage: complex 16x16 @ 16x256 via V_WMMA_F32_16X16X4_F32, in-place
// (each wave owns 2 disjoint 16-column tiles). CNOT chain = LDS permutation.
// ---------------------------------------------------------------------------
__global__ void __launch_bounds__(256) qc_main(
    const float* __restrict__ x,        // (4096, 12)
    const float* __restrict__ Mre,
    const float* __restrict__ Mim,
    float* __restrict__ out) {          // (4096, 12)
  extern __shared__ float smem[];
  float* sRe = smem;                    // [DIM]
  float* sIm = smem + DIM;              // [DIM]
  float* enc = smem + 2 * DIM;          // [32]: cos/sin of half encoding angles

  const int samp = blockIdx.x;
  const int t    = threadIdx.x;
  const int lane = t & 31;
  const int wave = t >> 5;

  // --- encoding angles (RY on |0>) ---
  if (t < NQ) {
    float ssq = 1e-10f;
    #pragma unroll
    for (int q = 0; q < NQ; ++q) { float v = x[samp * NQ + q]; ssq += v * v; }
    const float nrm = fmaxf(sqrtf(ssq), 1e-8f);
    const float ang = x[samp * NQ + t] * (3.14159265358979323846f / nrm);
    enc[t]      = cosf(0.5f * ang);
    enc[NQ + t] = sinf(0.5f * ang);
  }
  __syncthreads();

  float hc[NQ], hs[NQ];
  #pragma unroll
  for (int q = 0; q < NQ; ++q) { hc[q] = enc[q]; hs[q] = enc[NQ + q]; }

  // --- initial product state (real) + static CNOT-chain permutation g ---
  int gperm[16];
  #pragma unroll
  for (int k = 0; k < 16; ++k) {
    const int i = (k << 8) | t;
    float amp = 1.0f;
    #pragma unroll
    for (int q = 0; q < NQ; ++q)
      amp *= ((i >> (11 - q)) & 1) ? hs[q] : hc[q];
    sRe[i] = amp;
    sIm[i] = 0.0f;
    // new[i] = old[g(i)]; chain CNOT(q,q+1), q=0..10 applied in order:
    // g = f_0( f_1( ... f_10(i) ... ) ), f_q flips bit(10-q) if bit(11-q) set
    int j = i;
    #pragma unroll
    for (int q = 10; q >= 0; --q)
      j ^= ((j >> (11 - q)) & 1) << (10 - q);
    gperm[k] = j;
  }
  __syncthreads();

  const int m16   = lane & 15;
  const int khalf = (lane >> 4) << 1;   // A/B operand K-offset per lane half
  const int mb    = (lane >> 4) << 3;   // D row base per lane half

  for (int l = 0; l < 7; ++l) {
    #pragma unroll
    for (int b = 0; b < 3; ++b) {
      const float* __restrict__ mre = Mre + (l * 3 + b) * 256;
      const float* __restrict__ mim = Mim + (l * 3 + b) * 256;
      // A operand: 16x16 block unitary, split into four 16x4 K-chunks.
      // ISA layout: lanes 0-15 -> M=lane, {v0:K0,v1:K1}; lanes 16-31 -> {K2,K3}.
      v2f aRe[4], aIm[4], aNim[4];
      #pragma unroll
      for (int kc = 0; kc < 4; ++kc) {
        const int k0 = (kc << 2) + khalf;
        aRe[kc][0] = mre[m16 * 16 + k0];
        aRe[kc][1] = mre[m16 * 16 + k0 + 1];
        aIm[kc][0] = mim[m16 * 16 + k0];
        aIm[kc][1] = mim[m16 * 16 + k0 + 1];
        aNim[kc] = -aIm[kc];            // for Dre -= Aim*Bim (f32 WMMA has no A-neg)
      }
      #pragma unroll
      for (int tt = 0; tt < 2; ++tt) {
        const int r = ((wave * 2 + tt) << 4) | m16;   // this lane's column
        // B operand: state tile, 4x16 per K-chunk (mirrors A lane split).
        v2f bRe[4], bIm[4];
        #pragma unroll
        for (int kc = 0; kc < 4; ++kc) {
          const int k0 = (kc << 2) + khalf;
          const int a0 = saddr(b, k0, r);
          const int a1 = saddr(b, k0 + 1, r);
          bRe[kc][0] = sRe[a0]; bRe[kc][1] = sRe[a1];
          bIm[kc][0] = sIm[a0]; bIm[kc][1] = sIm[a1];
        }
        v8f cRe = {0.f, 0.f, 0.f, 0.f, 0.f, 0.f, 0.f, 0.f};
        v8f cIm = {0.f, 0.f, 0.f, 0.f, 0.f, 0.f, 0.f, 0.f};
        #pragma unroll
        for (int kc = 0; kc < 4; ++kc) {
          cRe = wmma_f32(aRe[kc],  bRe[kc], cRe);   // +Ar*Br
          cRe = wmma_f32(aNim[kc], bIm[kc], cRe);   // -Ai*Bi
          cIm = wmma_f32(aRe[kc],  bIm[kc], cIm);   // +Ar*Bi
          cIm = wmma_f32(aIm[kc],  bRe[kc], cIm);   // +Ai*Br
        }
        // D layout: VGPR v -> row mb+v, col = lane&15. In-place store is safe:
        // this wave's tiles own these columns exclusively, and its DS reads
        // complete (dscnt wait on WMMA operands) before the stores issue.
        #pragma unroll
        for (int v = 0; v < 8; ++v) {
          const int ad = saddr(b, mb + v, r);
          sRe[ad] = cRe[v];
          sIm[ad] = cIm[v];
        }
      }
      __syncthreads();
    }
    if (l < NLAYERS) {
      // full 11-CNOT chain = one static LDS permutation (register-staged)
      float pre[16], pim[16];
      #pragma unroll
      for (int k = 0; k < 16; ++k) { pre[k] = sRe[gperm[k]]; pim[k] = sIm[gperm[k]]; }
      __syncthreads();
      #pragma unroll
      for (int k = 0; k < 16; ++k) {
        const int i = (k << 8) | t;
        sRe[i] = pre[k]; sIm[i] = pim[k];
      }
      __syncthreads();
    }
  }

  // --- measurement: <Z_q> = sum_i |amp_i|^2 * (1 - 2*bit_q(i)) ---
  float z[NQ];
  #pragma unroll
  for (int q = 0; q < NQ; ++q) z[q] = 0.0f;
  #pragma unroll
  for (int k = 0; k < 16; ++k) {
    const int i = (k << 8) | t;
    const float re = sRe[i], im = sIm[i];
    const float p = re * re + im * im;
    #pragma unroll
    for (int q = 0; q < NQ; ++q)
      z[q] += ((i >> (11 - q)) & 1) ? -p : p;
  }
  __syncthreads();
  #pragma unroll
  for (int q = 0; q < NQ; ++q) sRe[q * 256 + t] = z[q];  // reuse state LDS
  __syncthreads();
  if (t < NQ) {
    float s = 0.0f;
    for (int j = 0; j < 256; ++j) s += sRe[t * 256 + j];
    out[samp * NQ + t] = s;
  }
}

// ---------------------------------------------------------------------------
extern "C" void kernel_launch(void* const* d_in, const int* in_sizes, int n_in,
                              void* d_out, int out_size, void* d_ws, size_t ws_size,
                              hipStream_t stream) {
  (void)in_sizes; (void)n_in; (void)out_size; (void)ws_size;
  const float* x  = (const float*)d_in[0];   // (4096,12)
  const float* w  = (const float*)d_in[1];   // (228,)
  const float* sc = (const float*)d_in[2];   // (6,)
  float* out = (float*)d_out;                // (4096,12)
  float* Mre = (float*)d_ws;                 // 21*256 f32
  float* Mim = Mre + NMAT * 256;             // 21*256 f32   (43KB total)

  qc_prep<<<NMAT, 256, 0, stream>>>(w, sc, Mre, Mim);

  const size_t shmem = (size_t)(2 * DIM + 32) * sizeof(float);  // ~32.1 KB
  qc_main<<<4096, 256, shmem, stream>>>(x, Mre, Mim, out);
}